// Model2_65077344469420
// MI455X (gfx1250) — compile-verified
//
#include <hip/hip_runtime.h>
#include <hip/hip_bf16.h>

typedef float v2f __attribute__((ext_vector_type(2)));
typedef float v8f __attribute__((ext_vector_type(8)));

#define NN    128
#define ROI2M 160
#define NF2   12720
#define CBTM  35
#define NF1   595
#define INTER 2048
#define CHN   1024

// ---------------------------------------------------------------------------
// C[128,Nc] = A[128,K] @ W[K,Nc]   (fp32 WMMA 16x16x4, full precision)
// Block = 256 thr (8 waves) covering 128(M) x 128(N); each wave: 16 N cols,
// all 8 M tiles (8 x v8f accumulators). A K-chunk (128x32) staged in LDS
// (row stride 36 floats: 16B-multiple rows -> b128 stores, conflict-free b64
// fragment reads). Hot loop is fully branch-free: out-of-range columns/rows
// are CLAMPED (not predicated) -- clamped lanes multiply against zero-filled
// A entries or land in never-stored columns, so results are unaffected.
// K%32 tail chunk is peeled.
// ---------------------------------------------------------------------------
__global__ __launch_bounds__(256) void gemm128_wmma(
    const float* __restrict__ A, const float* __restrict__ W,
    float* __restrict__ C, int K, int Nc) {
  __shared__ float As[128][36];   // [row][k within chunk], pad 36

  const int tid  = threadIdx.x;
  const int lane = tid & 31;
  const int wave = tid >> 5;
  const int half = lane >> 4;     // 0: K=k,k+1 ; 1: K=k+2,k+3 (A frag halves)
  const int l16  = lane & 15;
  const int n    = blockIdx.x * 128 + wave * 16 + l16;
  const bool nok = (n < Nc);
  const int  ncl = nok ? n : 0;   // clamped column -> always-safe addresses

  // staging coordinates: 128 rows x 32 k, 4 consecutive k per thread, 4 rounds
  const int srow0 = (tid * 4) >> 5;      // row for round 0 (rounds add 32)
  const int skk   = (tid * 4) & 31;      // k offset within chunk

  v8f acc[8] = {};                // 8 M-tiles of 16x16 f32 accum

  const int kfull = K & ~31;

  // running B pointer: row (half*2), clamped column; advances 32 rows/chunk
  const float* wp = W + (size_t)(half * 2) * Nc + ncl;
  const size_t wstep = (size_t)32 * Nc;

  for (int kc = 0; kc < kfull; kc += 32) {
    // branch-free cooperative staging of A[0:128, kc:kc+32]
#pragma unroll
    for (int it = 0; it < 4; ++it) {
      int r = srow0 + it * 32;
      const float* ap = A + (size_t)r * K + kc + skk;
      float4 t4;
      t4.x = ap[0]; t4.y = ap[1]; t4.z = ap[2]; t4.w = ap[3];
      *(float4*)&As[r][skk] = t4;        // 16B-aligned -> ds_store_b128
    }
    __syncthreads();

#pragma unroll
    for (int kk = 0; kk < 32; kk += 4) {
      v2f b;
      b.x = wp[(size_t)kk * Nc];          // rows kc+kk+half*2, kc+kk+half*2+1
      b.y = wp[(size_t)kk * Nc + Nc];
#pragma unroll
      for (int mt = 0; mt < 8; ++mt) {
        v2f a = *(const v2f*)&As[mt * 16 + l16][kk + half * 2]; // ds_load_b64
        acc[mt] = __builtin_amdgcn_wmma_f32_16x16x4_f32(
            false, a, false, b, (short)0, acc[mt], false, false);
      }
    }
    __syncthreads();
    wp += wstep;
  }

  if (kfull < K) {                 // peeled tail chunk (runs at most once)
#pragma unroll
    for (int it = 0; it < 4; ++it) {
      int r = srow0 + it * 32;
      float4 t4;
      t4.x = (kfull + skk + 0 < K) ? A[(size_t)r * K + kfull + skk + 0] : 0.0f;
      t4.y = (kfull + skk + 1 < K) ? A[(size_t)r * K + kfull + skk + 1] : 0.0f;
      t4.z = (kfull + skk + 2 < K) ? A[(size_t)r * K + kfull + skk + 2] : 0.0f;
      t4.w = (kfull + skk + 3 < K) ? A[(size_t)r * K + kfull + skk + 3] : 0.0f;
      *(float4*)&As[r][skk] = t4;
    }
    __syncthreads();
#pragma unroll
    for (int kk = 0; kk < 32; kk += 4) {
      const int k0 = kfull + kk + half * 2;
      // clamp rows to K-1: matching As entries are zero, contribution is 0
      const int kr0 = (k0     < K) ? k0     : K - 1;
      const int kr1 = (k0 + 1 < K) ? k0 + 1 : K - 1;
      v2f b;
      b.x = W[(size_t)kr0 * Nc + ncl];
      b.y = W[(size_t)kr1 * Nc + ncl];
#pragma unroll
      for (int mt = 0; mt < 8; ++mt) {
        v2f a = *(const v2f*)&As[mt * 16 + l16][kk + half * 2];
        acc[mt] = __builtin_amdgcn_wmma_f32_16x16x4_f32(
            false, a, false, b, (short)0, acc[mt], false, false);
      }
    }
    __syncthreads();
  }

  if (nok) {
#pragma unroll
    for (int mt = 0; mt < 8; ++mt)
#pragma unroll
      for (int r = 0; r < 8; ++r) {
        int row = mt * 16 + half * 8 + r;   // C/D layout: lanes 0-15 M=r, 16-31 M=8+r
        C[(size_t)row * Nc + n] = acc[mt][r];
      }
  }
}

// ---------------------------------------------------------------------------
__global__ void zero_f32(float* __restrict__ p, int n) {
  int i = blockIdx.x * blockDim.x + threadIdx.x;
  if (i < n) p[i] = 0.0f;
}

// degree over concat(dst, self-loops)
__global__ void deg_kernel(const int* __restrict__ dst, int E, int Nn,
                           float* __restrict__ deg) {
  int t = blockIdx.x * blockDim.x + threadIdx.x;
  if (t < E)           atomicAdd(&deg[dst[t]], 1.0f);
  else if (t < E + Nn) atomicAdd(&deg[t - E], 1.0f);
}

// out[d,:] += h[s,:] * (deg[s]^-1/2 * deg[d]^-1/2) over edges + self loops
__global__ void gcn_scatter(const float* __restrict__ h,
                            const int* __restrict__ src,
                            const int* __restrict__ dst,
                            int E, int Nn, const float* __restrict__ deg,
                            float* __restrict__ out, int F) {
  int tid = blockIdx.x * blockDim.x + threadIdx.x;
  int total = (E + Nn) * F;
  if (tid >= total) return;
  int e = tid / F;
  int j = tid - e * F;
  int s, d;
  if (e < E) { s = src[e]; d = dst[e]; } else { s = e - E; d = s; }
  float nrm = rsqrtf(fmaxf(deg[s], 1.0f)) * rsqrtf(fmaxf(deg[d], 1.0f));
  atomicAdd(&out[(size_t)d * F + j], h[(size_t)s * F + j] * nrm);
}

__global__ void bias_sigmoid(const float* __restrict__ in,
                             const float* __restrict__ b,
                             float* __restrict__ out, int total, int F) {
  int i = blockIdx.x * blockDim.x + threadIdx.x;
  if (i >= total) return;
  float t = in[i] + b[i % F];
  out[i] = 1.0f / (1.0f + __expf(-t));
}

// out[n] += sum_{k in chunk} v[k] * W[k,n]   (split-K vec-mat, out pre-zeroed)
__global__ void vecmat_part(const float* __restrict__ v,
                            const float* __restrict__ W,
                            float* __restrict__ out, int K, int Nc, int kChunk) {
  int n = blockIdx.x * blockDim.x + threadIdx.x;
  if (n >= Nc) return;
  int k0 = blockIdx.y * kChunk;
  int k1 = k0 + kChunk; if (k1 > K) k1 = K;
  float acc = 0.0f;
  for (int k = k0; k < k1; ++k)
    acc = fmaf(v[k], W[(size_t)k * Nc + n], acc);
  atomicAdd(&out[n], acc);
}

// [B, m(m-1)/2] -> [B, m, m] symmetric, zero diagonal (triu row-major order)
__global__ void anti_vec(const float* __restrict__ v, float* __restrict__ out,
                         int B, int m, int F) {
  int idx = blockIdx.x * blockDim.x + threadIdx.x;
  int total = B * m * m;
  if (idx >= total) return;
  int b = idx / (m * m);
  int r = idx - b * m * m;
  int i = r / m;
  int j = r - i * m;
  float val = 0.0f;
  if (i != j) {
    int a = i < j ? i : j;
    int c = i < j ? j : i;
    int p = a * m - (a * (a + 1)) / 2 + (c - a - 1);
    val = v[(size_t)b * F + p];
  }
  out[idx] = val;
}

// ---------------------------------------------------------------------------
extern "C" void kernel_launch(void* const* d_in, const int* in_sizes, int n_in,
                              void* d_out, int out_size, void* d_ws, size_t ws_size,
                              hipStream_t stream) {
  const float* x   = (const float*)d_in[0];
  const int*   ei  = (const int*)  d_in[1];
  const float* W23 = (const float*)d_in[2];  const float* b23 = (const float*)d_in[3];
  const float* W2  = (const float*)d_in[4];  const float* b2  = (const float*)d_in[5];
  const float* Wc1 = (const float*)d_in[6];  const float* bc1 = (const float*)d_in[7];
  const float* W24 = (const float*)d_in[8];  const float* b24 = (const float*)d_in[9];
  const float* W25 = (const float*)d_in[10]; const float* b25 = (const float*)d_in[11];
  const float* W26 = (const float*)d_in[12]; const float* b26 = (const float*)d_in[13];
  const float* Wl1 = (const float*)d_in[14]; const float* bl1 = (const float*)d_in[15];
  const float* Wl2 = (const float*)d_in[16]; const float* bl2 = (const float*)d_in[17];

  float* out = (float*)d_out;
  float* ws  = (float*)d_ws;

  const int E = in_sizes[1] / 2;
  const int* src = ei;
  const int* dst = ei + E;

  // workspace layout (floats)
  size_t oh   = 0;                       // gemm result     128*NF2
  size_t og   = oh   + (size_t)NN * NF2; // aggregation     128*NF2
  size_t oy3  = og   + (size_t)NN * NF2; // y3              128*NF2
  size_t ox1  = oy3  + (size_t)NN * NF2; // x1              128*INTER
  size_t oy2  = ox1  + (size_t)NN * INTER;
  size_t ox2  = oy2  + (size_t)NN * INTER; // x2            128*NF1
  size_t oz   = ox2  + (size_t)NN * NF1;
  size_t oy1  = oz   + (size_t)NN * NF1;
  size_t odeg = oy1  + (size_t)NN * NF1;  // 128
  size_t oi1p = odeg + 128;               // 1024
  size_t ocbp = oi1p + CHN;               // 595
  size_t ocbt = ocbp + NF1;               // 595

  // output layout (floats)
  const size_t OFF_XM  = 0;                                   // 128*160*160
  const size_t OFF_ZM  = OFF_XM + (size_t)NN * ROI2M * ROI2M; // 128*35*35
  const size_t OFF_YM  = OFF_ZM + (size_t)NN * CBTM * CBTM;
  const size_t OFF_I1  = OFF_YM + (size_t)NN * ROI2M * ROI2M; // 1024
  const size_t OFF_CBT = OFF_I1 + CHN;                        // 35*35

  // degree (shared by all GCN layers)
  zero_f32<<<1, 128, 0, stream>>>(ws + odeg, NN);
  deg_kernel<<<(E + NN + 255) / 256, 256, 0, stream>>>(dst, E, NN, ws + odeg);

  auto gcnL = [&](const float* Ain, const float* Wt, const float* bias,
                  float* dstA, int K, int F) {
    gemm128_wmma<<<dim3((F + 127) / 128), 256, 0, stream>>>(Ain, Wt, ws + oh, K, F);
    int tot = NN * F;
    zero_f32<<<(tot + 255) / 256, 256, 0, stream>>>(ws + og, tot);
    int totE = (E + NN) * F;
    gcn_scatter<<<(totE + 255) / 256, 256, 0, stream>>>(
        ws + oh, src, dst, E, NN, ws + odeg, ws + og, F);
    bias_sigmoid<<<(tot + 255) / 256, 256, 0, stream>>>(ws + og, bias, dstA, tot, F);
  };

  // encoder
  gcnL(x,        W23, b23, ws + ox1, NF2,   INTER);
  gcnL(ws + ox1, W2,  b2,  ws + ox2, INTER, NF1);
  gcnL(ws + ox2, Wc1, bc1, ws + oz,  NF1,   NF1);

  // normalizer: i1 = sig(zf @ Wl1 + bl1) ; cbt = sig(i1 @ Wl2 + bl2)
  zero_f32<<<(CHN + 255) / 256, 256, 0, stream>>>(ws + oi1p, CHN);
  {
    int K = NN * NF1;                 // 76160
    int kChunk = 8192;
    dim3 g((CHN + 255) / 256, (K + kChunk - 1) / kChunk);
    vecmat_part<<<g, 256, 0, stream>>>(ws + oz, Wl1, ws + oi1p, K, CHN, kChunk);
  }
  bias_sigmoid<<<(CHN + 255) / 256, 256, 0, stream>>>(ws + oi1p, bl1,
                                                      out + OFF_I1, CHN, CHN);
  zero_f32<<<(NF1 + 255) / 256, 256, 0, stream>>>(ws + ocbp, NF1);
  vecmat_part<<<dim3((NF1 + 255) / 256, 1), 256, 0, stream>>>(
      out + OFF_I1, Wl2, ws + ocbp, CHN, NF1, CHN);
  bias_sigmoid<<<(NF1 + 255) / 256, 256, 0, stream>>>(ws + ocbp, bl2,
                                                      ws + ocbt, NF1, NF1);

  // decoder
  gcnL(ws + oz,  W24, b24, ws + oy1, NF1,   NF1);
  gcnL(ws + oy1, W25, b25, ws + oy2, NF1,   INTER);
  gcnL(ws + oy2, W26, b26, ws + oy3, INTER, NF2);

  // anti-vectorize outputs
  {
    int t1 = NN * ROI2M * ROI2M;
    anti_vec<<<(t1 + 255) / 256, 256, 0, stream>>>(x,        out + OFF_XM, NN, ROI2M, NF2);
    anti_vec<<<(t1 + 255) / 256, 256, 0, stream>>>(ws + oy3, out + OFF_YM, NN, ROI2M, NF2);
    int t2 = NN * CBTM * CBTM;
    anti_vec<<<(t2 + 255) / 256, 256, 0, stream>>>(ws + oz,  out + OFF_ZM, NN, CBTM, NF1);
    int t3 = CBTM * CBTM;
    anti_vec<<<(t3 + 255) / 256, 256, 0, stream>>>(ws + ocbt, out + OFF_CBT, 1, CBTM, NF1);
  }
}